// Net_47579647705716
// MI455X (gfx1250) — compile-verified
//
#include <hip/hip_runtime.h>
#include <math.h>

typedef __attribute__((ext_vector_type(2))) float v2f;
typedef __attribute__((ext_vector_type(8))) float v8f;

#define IN_C 128
#define NEG_SLOPE 0.2f

static inline int cdiv(long a, long b) { return (int)((a + b - 1) / b); }

// ---- order-preserving float<->uint encoding for atomic max on f32 ----
__device__ __forceinline__ unsigned enc_f32(float f) {
    unsigned u = __float_as_uint(f);
    return (u & 0x80000000u) ? ~u : (u | 0x80000000u);
}
__device__ __forceinline__ float dec_f32(unsigned e) {
    unsigned u = (e & 0x80000000u) ? (e ^ 0x80000000u) : ~e;
    return __uint_as_float(u);
}
#define ENC_NEG_INF 0x007FFFFFu   // enc(-inf)

// ------------------------- fill / pad kernels -------------------------
__global__ void fill_u32(unsigned* __restrict__ p, unsigned v, int n) {
    int i = blockIdx.x * blockDim.x + threadIdx.x;
    if (i < n) p[i] = v;
}
__global__ void fill_f32(float* __restrict__ p, float v, int n) {
    int i = blockIdx.x * blockDim.x + threadIdx.x;
    if (i < n) p[i] = v;
}
template<int F>
__global__ void fill_bias(float* __restrict__ p, const float* __restrict__ b, int n) {
    int i = blockIdx.x * blockDim.x + threadIdx.x;
    if (i < n) p[i] = b[i % F];
}
// Wp[K x Fp] = W[K x F] zero-padded in columns
__global__ void pad_cols(const float* __restrict__ W, float* __restrict__ Wp,
                         int K, int F, int Fp) {
    int i = blockIdx.x * blockDim.x + threadIdx.x;
    if (i >= K * Fp) return;
    int r = i / Fp, c = i % Fp;
    Wp[i] = (c < F) ? W[r * F + c] : 0.f;
}

// ------------------------- WMMA GEMM (f32, 16x16x4) -------------------------
// H[M x F] = A[M x K] @ B[K x F]. One wave32 per 16x16 tile, fully unrolled,
// no guards: M % 16 == 0 (100000 = 6250*16), K % 4 == 0, F % 16 == 0 (padded).
template<int K, int F>
__global__ void gemm_wmma_f32(const float* __restrict__ A, const float* __restrict__ B,
                              float* __restrict__ Hout) {
    const int lane = threadIdx.x;        // 0..31
    const int half = lane >> 4;          // 0 or 1
    const int l    = lane & 15;
    const int row  = blockIdx.x * 16 + l;
    const int col  = blockIdx.y * 16 + l;

    v8f c = {0.f, 0.f, 0.f, 0.f, 0.f, 0.f, 0.f, 0.f};
    const float* __restrict__ arow = A + (size_t)row * K + half * 2;  // 8B aligned
    const float* __restrict__ bcol = B + (size_t)(half * 2) * F + col;

    #pragma unroll
    for (int k0 = 0; k0 < K; k0 += 4) {
        // A 16x4 fragment: lanes 0-15 -> K=k0,k0+1 ; lanes 16-31 -> K=k0+2,k0+3
        v2f a = *(const v2f*)(arow + k0);       // one global_load_b64
        v2f b;
        b.x = bcol[(size_t)k0 * F];
        b.y = bcol[(size_t)(k0 + 1) * F];
        c = __builtin_amdgcn_wmma_f32_16x16x4_f32(false, a, false, b, (short)0, c,
                                                  false, false);
    }

    const int mbase = blockIdx.x * 16 + half * 8;   // C/D: VGPR v -> row v + 8*half
    #pragma unroll
    for (int v = 0; v < 8; ++v)
        Hout[(size_t)(mbase + v) * F + col] = c[v];
}

// ------------------------- per-node attention scores -------------------------
// esrc[n,h] = sum_c hbuf[n*LDH + h*C + c] * a_src[h*C + c] ; edst analogous.
template<int Hh, int C, int LDH>
__global__ void node_scores(const float* __restrict__ hbuf,
                            const float* __restrict__ a_src, const float* __restrict__ a_dst,
                            float* __restrict__ esrc, float* __restrict__ edst, int Nn) {
    int i = blockIdx.x * blockDim.x + threadIdx.x;
    if (i >= Nn * Hh) return;
    int n = i / Hh, hd = i % Hh;
    const float* __restrict__ hp = hbuf + (size_t)n * LDH + hd * C;
    const float* __restrict__ as = a_src + hd * C;
    const float* __restrict__ ad = a_dst + hd * C;
    float s = 0.f, d = 0.f;
    #pragma unroll
    for (int c = 0; c < C; ++c) { float v = hp[c]; s += v * as[c]; d += v * ad[c]; }
    esrc[i] = s;
    edst[i] = d;
}

// ------------------------- edge passes -------------------------
__device__ __forceinline__ void edge_nodes(const int* __restrict__ src, const int* __restrict__ dst,
                                           int e, int E, int& s, int& d) {
    if (e < E) { s = src[e]; d = dst[e]; }
    else       { s = e - E; d = e - E; }     // self loops appended
}

template<int Hh>
__device__ __forceinline__ float edge_e(const float* __restrict__ esrc,
                                        const float* __restrict__ edst,
                                        int s, int d, int hd) {
    float v = esrc[s * Hh + hd] + edst[d * Hh + hd];
    return (v > 0.f) ? v : NEG_SLOPE * v;
}

// pass A: segment max over dst
template<int Hh>
__global__ void edge_max(const int* __restrict__ src, const int* __restrict__ dst,
                         const float* __restrict__ esrc, const float* __restrict__ edst,
                         unsigned* __restrict__ m, int E, int Etot) {
    int tid = blockIdx.x * blockDim.x + threadIdx.x;
    if (tid >= Etot * Hh) return;
    int e = tid / Hh, hd = tid % Hh;
    int s, d; edge_nodes(src, dst, e, E, s, d);
    float v = edge_e<Hh>(esrc, edst, s, d, hd);
    atomicMax(&m[d * Hh + hd], enc_f32(v));
}

// pass B: segment sum of exp(e - m)
template<int Hh>
__global__ void edge_sum(const int* __restrict__ src, const int* __restrict__ dst,
                         const float* __restrict__ esrc, const float* __restrict__ edst,
                         const unsigned* __restrict__ m, float* __restrict__ ssum,
                         int E, int Etot) {
    int tid = blockIdx.x * blockDim.x + threadIdx.x;
    if (tid >= Etot * Hh) return;
    int e = tid / Hh, hd = tid % Hh;
    int s, d; edge_nodes(src, dst, e, E, s, d);
    float v = edge_e<Hh>(esrc, edst, s, d, hd);
    atomicAdd(&ssum[d * Hh + hd], __expf(v - dec_f32(m[d * Hh + hd])));
}

// pass C: agg[d*F + f] += alpha[e, f/C] * hbuf[s*LDH + f]  (thread per edge-feature)
template<int Hh, int C, int LDH>
__global__ void edge_aggregate(const int* __restrict__ src, const int* __restrict__ dst,
                               const float* __restrict__ esrc, const float* __restrict__ edst,
                               const unsigned* __restrict__ m, const float* __restrict__ ssum,
                               const float* __restrict__ hbuf, float* __restrict__ agg,
                               int E, int Etot) {
    constexpr int F = Hh * C;
    int tid = blockIdx.x * blockDim.x + threadIdx.x;
    if (tid >= Etot * F) return;
    int e = tid / F, f = tid % F;
    int hd = f / C;
    int s, d; edge_nodes(src, dst, e, E, s, d);
    float v = edge_e<Hh>(esrc, edst, s, d, hd);
    int ih = d * Hh + hd;
    float alpha = __expf(v - dec_f32(m[ih])) / (ssum[ih] + 1e-16f);
    atomicAdd(&agg[(size_t)d * F + f], alpha * hbuf[(size_t)s * LDH + f]);
}

// ------------------------- ELU -------------------------
__global__ void elu_kernel(const float* __restrict__ in, float* __restrict__ out, int n) {
    int i = blockIdx.x * blockDim.x + threadIdx.x;
    if (i >= n) return;
    float v = in[i];
    out[i] = (v > 0.f) ? v : (__expf(v) - 1.f);
}

// ------------------------- one GAT layer -------------------------
// K: input features, Hh: heads, C: out per head, LDH: row stride of h buffer
// (LDH == Hh*C except layer 3, where W is padded to LDH columns).
template<int K, int Hh, int C, int LDH>
static void run_gat_layer(const float* x, const float* W /* K x LDH */,
                          const float* a_src, const float* a_dst, const float* bias,
                          float* hbuf, float* esrc, float* edst,
                          unsigned* mbuf, float* ssum, float* agg,
                          const int* srcp, const int* dstp, int Nn, int E, int Etot,
                          hipStream_t stream) {
    constexpr int F = Hh * C;
    static_assert(LDH % 16 == 0, "padded h stride must be multiple of 16");
    // h = x @ W
    dim3 gg(Nn / 16, LDH / 16);
    gemm_wmma_f32<K, LDH><<<gg, 32, 0, stream>>>(x, W, hbuf);
    // per-node scores
    int nh = Nn * Hh;
    node_scores<Hh, C, LDH><<<cdiv(nh, 256), 256, 0, stream>>>(hbuf, a_src, a_dst,
                                                               esrc, edst, Nn);
    // init reductions + bias prefill (out = segsum + bias)
    fill_u32<<<cdiv(nh, 256), 256, 0, stream>>>(mbuf, ENC_NEG_INF, nh);
    fill_f32<<<cdiv(nh, 256), 256, 0, stream>>>(ssum, 0.f, nh);
    fill_bias<F><<<cdiv((long)Nn * F, 256), 256, 0, stream>>>(agg, bias, Nn * F);
    // edge passes
    int eh = Etot * Hh;
    edge_max<Hh><<<cdiv(eh, 256), 256, 0, stream>>>(srcp, dstp, esrc, edst, mbuf, E, Etot);
    edge_sum<Hh><<<cdiv(eh, 256), 256, 0, stream>>>(srcp, dstp, esrc, edst, mbuf, ssum,
                                                    E, Etot);
    int ef = Etot * F;  // max 3.3M * 64 = 211.2M < 2^31
    edge_aggregate<Hh, C, LDH><<<cdiv(ef, 256), 256, 0, stream>>>(srcp, dstp, esrc, edst,
                                                                  mbuf, ssum, hbuf, agg,
                                                                  E, Etot);
}

extern "C" void kernel_launch(void* const* d_in, const int* in_sizes, int n_in,
                              void* d_out, int out_size, void* d_ws, size_t ws_size,
                              hipStream_t stream) {
    const float* x       = (const float*)d_in[0];
    const int*   ei      = (const int*)d_in[1];
    const float* W1      = (const float*)d_in[2];
    const float* a_src1  = (const float*)d_in[3];
    const float* a_dst1  = (const float*)d_in[4];
    const float* b1      = (const float*)d_in[5];
    const float* W2      = (const float*)d_in[6];
    const float* a_src2  = (const float*)d_in[7];
    const float* a_dst2  = (const float*)d_in[8];
    const float* b2      = (const float*)d_in[9];
    const float* W3      = (const float*)d_in[10];
    const float* a_src3  = (const float*)d_in[11];
    const float* a_dst3  = (const float*)d_in[12];
    const float* b3      = (const float*)d_in[13];

    const int N = in_sizes[0] / IN_C;     // 100000
    const int E = in_sizes[1] / 2;        // 3200000
    const int Etot = E + N;               // self loops appended implicitly
    const int* srcp = ei;
    const int* dstp = ei + E;

    // workspace layout (floats); total ~= N*240 + 3072 floats ~= 96 MB
    float* ws    = (float*)d_ws;
    float* x2    = ws;               ws += (size_t)N * 16;
    float* x3    = ws;               ws += (size_t)N * 64;
    float* hbuf  = ws;               ws += (size_t)N * 64;   // also fits N*48 (layer 3)
    float* esrc  = ws;               ws += (size_t)N * 8;
    float* edst  = ws;               ws += (size_t)N * 8;
    unsigned* mbuf = (unsigned*)ws;  ws += (size_t)N * 8;
    float* ssum  = ws;               ws += (size_t)N * 8;
    float* agg   = ws;               ws += (size_t)N * 64;
    float* W3p   = ws;               ws += (size_t)64 * 48;  // W3 padded 64x40 -> 64x48

    // ---- layer 1: 128 -> 4 heads x 4, concat -> 16, ELU ----
    run_gat_layer<IN_C, 4, 4, 16>(x, W1, a_src1, a_dst1, b1,
                                  hbuf, esrc, edst, mbuf, ssum, agg,
                                  srcp, dstp, N, E, Etot, stream);
    elu_kernel<<<cdiv((long)N * 16, 256), 256, 0, stream>>>(agg, x2, N * 16);

    // ---- layer 2: 16 -> 8 heads x 8, concat -> 64, ELU ----
    run_gat_layer<16, 8, 8, 64>(x2, W2, a_src2, a_dst2, b2,
                                hbuf, esrc, edst, mbuf, ssum, agg,
                                srcp, dstp, N, E, Etot, stream);
    elu_kernel<<<cdiv((long)N * 64, 256), 256, 0, stream>>>(agg, x3, N * 64);

    // ---- layer 3: 64 -> 1 head x 40, mean over 1 head == identity, no ELU ----
    // pad W3 columns 40 -> 48 so the GEMM runs guard-free; h stride = 48
    pad_cols<<<cdiv(64 * 48, 256), 256, 0, stream>>>(W3, W3p, 64, 40, 48);
    run_gat_layer<64, 1, 40, 48>(x3, W3p, a_src3, a_dst3, b3,
                                 hbuf, esrc, edst, mbuf, ssum, (float*)d_out,
                                 srcp, dstp, N, E, Etot, stream);
}